// DeeperGCN_38920993636380
// MI455X (gfx1250) — compile-verified
//
#include <hip/hip_runtime.h>
#include <hip/hip_bf16.h>
#include <float.h>

#define N_NODES 50000
#define N_EDGES 800000
#define IN_CH   128
#define HID     64
#define N_LAYERS 14
#define EPS_MSG 1e-7f
#define EPS_SM  1e-16f
#define LN_EPS  1e-5f

typedef float v2f __attribute__((ext_vector_type(2)));
typedef float v8f __attribute__((ext_vector_type(8)));

// Sign-aware float atomic max using integer atomics (no CAS loop).
// Works for mixed-sign values; init location to -FLT_MAX.
__device__ __forceinline__ void atomicMaxF(float* addr, float val) {
    if (val >= 0.0f) atomicMax((int*)addr, __float_as_int(val));
    else             atomicMin((unsigned int*)addr, __float_as_uint(val));
}

// ---------------------------------------------------------------------------
// Persistent-wave WMMA GEMM: C[ntiles*16, 64] = A[ntiles*16, K] * W[K, 64] + b
// blockDim = 128 (4 waves); wave w owns output columns [16w, 16w+16).
// B fragments for the whole K loop are preloaded into VGPRs per wave.
// Uses V_WMMA_F32_16X16X4_F32 (full fp32 precision).
// ---------------------------------------------------------------------------
template<int K>
__global__ void gemm_wmma(const float* __restrict__ A,
                          const float* __restrict__ W,
                          const float* __restrict__ bias,
                          float* __restrict__ C,
                          int accumulate, int ntiles) {
    const int wave = threadIdx.x >> 5;
    const int lane = threadIdx.x & 31;
    const int hi   = lane >> 4;   // 0: K-pair {0,1}; 1: K-pair {2,3}
    const int mn   = lane & 15;   // A: row-in-tile; B/C: col-in-tile
    const int colbase = wave * 16;
    constexpr int KSTEPS = K / 4;

    // Preload B fragments: b = { W[k0+2hi][col], W[k0+2hi+1][col] }
    v2f bf[KSTEPS];
#pragma unroll
    for (int k = 0; k < KSTEPS; ++k) {
        const int kk = 4 * k + 2 * hi;
        bf[k].x = W[(size_t)kk       * HID + colbase + mn];
        bf[k].y = W[(size_t)(kk + 1) * HID + colbase + mn];
    }
    const float bv = bias[colbase + mn];

    for (int tile = blockIdx.x; tile < ntiles; tile += gridDim.x) {
        const float* Arow = A + (size_t)(tile * 16 + mn) * K;
        // speculative prefetch of next tile's A row (dropped if OOB)
        __builtin_prefetch(Arow + (size_t)gridDim.x * 16 * K, 0, 0);
        v8f acc = {};
#pragma unroll
        for (int k = 0; k < KSTEPS; ++k) {
            const int kk = 4 * k + 2 * hi;
            v2f a;
            a.x = Arow[kk];
            a.y = Arow[kk + 1];
            acc = __builtin_amdgcn_wmma_f32_16x16x4_f32(
                /*neg_a=*/false, a, /*neg_b=*/false, bf[k],
                /*c_mod=*/(short)0, acc, /*reuse_a=*/false, /*reuse_b=*/false);
        }
        // C/D layout: VGPR v, lane hi*16+n  ->  element [M = v + 8*hi][N = n]
#pragma unroll
        for (int v = 0; v < 8; ++v) {
            const int r = tile * 16 + v + 8 * hi;
            const size_t idx = (size_t)r * HID + colbase + mn;
            const float val = acc[v] + bv;
            if (accumulate) C[idx] += val;
            else            C[idx] = val;
        }
    }
}

// ---------------------------------------------------------------------------
// Per-layer scratch init: m = -FLT_MAX, s = 0, w = 0
// ---------------------------------------------------------------------------
__global__ void seg_init(float* __restrict__ m, float* __restrict__ s,
                         float* __restrict__ w, int n) {
    int i = blockIdx.x * blockDim.x + threadIdx.x;
    if (i < n) { m[i] = -FLT_MAX; s[i] = 0.0f; w[i] = 0.0f; }
}

// ---------------------------------------------------------------------------
// Edge pass 1: per-(dst, ch) running max of t * (relu(y[src]) + eps)
// One thread per (edge, channel-pair); float2 gathers.
// ---------------------------------------------------------------------------
__global__ void edge_max(const float* __restrict__ y,
                         const int* __restrict__ src, const int* __restrict__ dst,
                         float* __restrict__ m,
                         const float* __restrict__ tarr, int L) {
    const long long gid = (long long)blockIdx.x * blockDim.x + threadIdx.x;
    const int e = (int)(gid >> 5);
    if (e >= N_EDGES) return;
    const int c = ((int)gid & 31) * 2;
    const float t = tarr[L];
    const int sN = src[e], dN = dst[e];
    const float* yp = y + (size_t)sN * HID + c;
    const float s0 = t * (fmaxf(yp[0], 0.0f) + EPS_MSG);
    const float s1 = t * (fmaxf(yp[1], 0.0f) + EPS_MSG);
    float* mp = m + (size_t)dN * HID + c;
    atomicMaxF(mp,     s0);
    atomicMaxF(mp + 1, s1);
}

// ---------------------------------------------------------------------------
// Edge pass 2 (fused): e = exp(score - m[dst]); s += e; w += msg * e
// The softmax denominator factors out of the weighted sum, so the normalized
// aggregation is recovered node-wise as w / (s + EPS_SM).
// ---------------------------------------------------------------------------
__global__ void edge_sum(const float* __restrict__ y,
                         const int* __restrict__ src, const int* __restrict__ dst,
                         const float* __restrict__ m,
                         float* __restrict__ s, float* __restrict__ w,
                         const float* __restrict__ tarr, int L) {
    const long long gid = (long long)blockIdx.x * blockDim.x + threadIdx.x;
    const int e = (int)(gid >> 5);
    if (e >= N_EDGES) return;
    const int c = ((int)gid & 31) * 2;
    const float t = tarr[L];
    const int sN = src[e], dN = dst[e];
    const float* yp = y + (size_t)sN * HID + c;
    const float msg0 = fmaxf(yp[0], 0.0f) + EPS_MSG;
    const float msg1 = fmaxf(yp[1], 0.0f) + EPS_MSG;
    const size_t off = (size_t)dN * HID + c;
    const float e0 = __expf(t * msg0 - m[off]);
    const float e1 = __expf(t * msg1 - m[off + 1]);
    atomicAdd(s + off,     e0);
    atomicAdd(s + off + 1, e1);
    atomicAdd(w + off,     msg0 * e0);
    atomicAdd(w + off + 1, msg1 * e1);
}

// ---------------------------------------------------------------------------
// Node-wise: z = aggr + y = w/(s+eps) + y   (empty segments -> aggr == 0)
// ---------------------------------------------------------------------------
__global__ void node_combine(const float* __restrict__ y,
                             const float* __restrict__ s,
                             const float* __restrict__ w,
                             float* __restrict__ z, int n) {
    int i = blockIdx.x * blockDim.x + threadIdx.x;
    if (i < n) z[i] = w[i] / (s[i] + EPS_SM) + y[i];
}

// ---------------------------------------------------------------------------
// y = relu(layernorm(h)); one wave32 per node (64 channels, 2 per lane)
// ---------------------------------------------------------------------------
__global__ void ln_relu(const float* __restrict__ h, float* __restrict__ y) {
    const int tid = blockIdx.x * blockDim.x + threadIdx.x;
    const int node = tid >> 5;
    const int lane = tid & 31;
    if (node >= N_NODES) return;
    const float* hp = h + (size_t)node * HID + lane * 2;
    const float a = hp[0], b = hp[1];
    float sum = a + b;
#pragma unroll
    for (int off = 16; off > 0; off >>= 1) sum += __shfl_xor(sum, off, 32);
    const float mu = sum * (1.0f / HID);
    const float da = a - mu, db = b - mu;
    float var = da * da + db * db;
#pragma unroll
    for (int off = 16; off > 0; off >>= 1) var += __shfl_xor(var, off, 32);
    const float r = rsqrtf(var * (1.0f / HID) + LN_EPS);
    float* yp = y + (size_t)node * HID + lane * 2;
    yp[0] = fmaxf(da * r, 0.0f);
    yp[1] = fmaxf(db * r, 0.0f);
}

// ---------------------------------------------------------------------------
// out = log_softmax(relu(layernorm(h))); one wave32 per node
// ---------------------------------------------------------------------------
__global__ void final_lsm(const float* __restrict__ h, float* __restrict__ out) {
    const int tid = blockIdx.x * blockDim.x + threadIdx.x;
    const int node = tid >> 5;
    const int lane = tid & 31;
    if (node >= N_NODES) return;
    const float* hp = h + (size_t)node * HID + lane * 2;
    const float a = hp[0], b = hp[1];
    float sum = a + b;
#pragma unroll
    for (int off = 16; off > 0; off >>= 1) sum += __shfl_xor(sum, off, 32);
    const float mu = sum * (1.0f / HID);
    const float da = a - mu, db = b - mu;
    float var = da * da + db * db;
#pragma unroll
    for (int off = 16; off > 0; off >>= 1) var += __shfl_xor(var, off, 32);
    const float r = rsqrtf(var * (1.0f / HID) + LN_EPS);
    const float oa = fmaxf(da * r, 0.0f);
    const float ob = fmaxf(db * r, 0.0f);
    float mx = fmaxf(oa, ob);
#pragma unroll
    for (int off = 16; off > 0; off >>= 1) mx = fmaxf(mx, __shfl_xor(mx, off, 32));
    float se = __expf(oa - mx) + __expf(ob - mx);
#pragma unroll
    for (int off = 16; off > 0; off >>= 1) se += __shfl_xor(se, off, 32);
    const float ls = mx + logf(se);
    float* op = out + (size_t)node * HID + lane * 2;
    op[0] = oa - ls;
    op[1] = ob - ls;
}

// ---------------------------------------------------------------------------
extern "C" void kernel_launch(void* const* d_in, const int* in_sizes, int n_in,
                              void* d_out, int out_size, void* d_ws, size_t ws_size,
                              hipStream_t stream) {
    const float* x    = (const float*)d_in[0];
    const int*   eidx = (const int*)d_in[1];
    const float* encW = (const float*)d_in[2];
    const float* encb = (const float*)d_in[3];
    const float* mlpW = (const float*)d_in[4];
    const float* mlpb = (const float*)d_in[5];
    const float* tarr = (const float*)d_in[6];
    const int* src = eidx;
    const int* dst = eidx + N_EDGES;
    float* out = (float*)d_out;

    const size_t NV = (size_t)N_NODES * HID;
    float* h = (float*)d_ws;      // current node features
    float* y = h + NV;            // relu(layernorm(h))
    float* m = y + NV;            // segment max (reused as z = aggr + y)
    float* s = m + NV;            // sum of exp
    float* w = s + NV;            // sum of msg*exp
    float* z = m;                 // alias: m is dead once edge_sum is done

    constexpr int NTILES = N_NODES / 16;           // 3125 (exact)
    const int nvals = (int)NV;
    const long long ethreads = (long long)N_EDGES * 32;
    const int eblocks = (int)((ethreads + 255) / 256);
    const int nblocks = (nvals + 255) / 256;
    const int wblocks = (N_NODES * 32 + 255) / 256;

    // Encoder: h = x @ enc_W + enc_b   (K = 128)
    gemm_wmma<IN_CH><<<512, 128, 0, stream>>>(x, encW, encb, h, 0, NTILES);

    for (int L = 0; L < N_LAYERS; ++L) {
        const float* yin = (L == 0) ? h : y;
        if (L > 0) ln_relu<<<wblocks, 256, 0, stream>>>(h, y);
        seg_init<<<nblocks, 256, 0, stream>>>(m, s, w, nvals);
        edge_max<<<eblocks, 256, 0, stream>>>(yin, src, dst, m, tarr, L);
        edge_sum<<<eblocks, 256, 0, stream>>>(yin, src, dst, m, s, w, tarr, L);
        node_combine<<<nblocks, 256, 0, stream>>>(yin, s, w, z, nvals);
        // layer 0 overwrites h; layers >= 1 accumulate (res+ carry)
        gemm_wmma<HID><<<512, 128, 0, stream>>>(z, mlpW + (size_t)L * HID * HID,
                                                mlpb + (size_t)L * HID, h,
                                                (L > 0) ? 1 : 0, NTILES);
    }

    final_lsm<<<wblocks, 256, 0, stream>>>(h, out);
}